// DenseContrastiveLoss_24902220382316
// MI455X (gfx1250) — compile-verified
//
#include <hip/hip_runtime.h>
#include <math.h>

#define Bb 8
#define Cc 256
#define Hh 128
#define Ww 128
#define Nn (Hh*Ww)
#define NRAND 32
#define NHARD 96
#define NPOS 256
#define INVT (1.0f/0.07f)
#define APAD 260

typedef __attribute__((ext_vector_type(2))) float v2f;
typedef __attribute__((ext_vector_type(8))) float v8f;

// ---------------- Kernel 1: per-pixel inverse L2 norms ----------------
__global__ void knorm(const float* __restrict__ rgb, const float* __restrict__ dep,
                      float* __restrict__ inv_rgb, float* __restrict__ inv_dep) {
    int e = blockIdx.x * blockDim.x + threadIdx.x;   // 0..B*N-1
    int b = e / Nn, n = e % Nn;
    const float* rp = rgb + (size_t)b * Cc * Nn + n;
    const float* dp = dep + (size_t)b * Cc * Nn + n;
    float sr = 0.f, sd = 0.f;
    for (int c = 0; c < Cc; ++c) {
        float r = rp[(size_t)c * Nn];
        float d = dp[(size_t)c * Nn];
        sr = fmaf(r, r, sr);
        sd = fmaf(d, d, sd);
    }
    inv_rgb[e] = 1.0f / fmaxf(sqrtf(sr), 1e-12f);
    inv_dep[e] = 1.0f / fmaxf(sqrtf(sd), 1e-12f);
}

// ---------------- Kernel 2: pos_sim (one wave per pixel, bilinear gather) ----------------
__global__ void kpos(const float* __restrict__ rgb, const float* __restrict__ dep,
                     const float* __restrict__ pc,
                     const float* __restrict__ inv_rgb, const float* __restrict__ inv_dep,
                     float* __restrict__ pos_sim) {
    int wave = threadIdx.x >> 5, lane = threadIdx.x & 31;
    int g = blockIdx.x * 8 + wave;                  // pixel id in [0, B*N)
    int b = g / Nn, n = g % Nn;
    float pu = pc[((size_t)b * 2 + 0) * Nn + n];
    float pv = pc[((size_t)b * 2 + 1) * Nn + n];
    float x0 = floorf(pu), y0 = floorf(pv);
    float wx = pu - x0, wy = pv - y0;
    int x0i = (int)x0, y0i = (int)y0;
    float wgt[4] = {(1.f-wx)*(1.f-wy), wx*(1.f-wy), (1.f-wx)*wy, wx*wy};
    int xs[4] = {x0i, x0i+1, x0i,   x0i+1};
    int ys[4] = {y0i, y0i,   y0i+1, y0i+1};
    int idx[4]; float cf[4];
    #pragma unroll
    for (int k = 0; k < 4; ++k) {
        bool valid = (xs[k] >= 0) && (xs[k] < Ww) && (ys[k] >= 0) && (ys[k] < Hh);
        int ix = min(max(xs[k], 0), Ww-1);
        int iy = min(max(ys[k], 0), Hh-1);
        idx[k] = iy * Ww + ix;
        cf[k] = valid ? wgt[k] * inv_rgb[(size_t)b * Nn + idx[k]] : 0.f;
    }
    const float* rp = rgb + (size_t)b * Cc * Nn;
    const float* dp = dep + (size_t)b * Cc * Nn;
    float acc = 0.f;
    #pragma unroll
    for (int i = 0; i < 8; ++i) {
        int c = lane + 32 * i;
        size_t base = (size_t)c * Nn;
        float r = cf[0]*rp[base+idx[0]] + cf[1]*rp[base+idx[1]]
                + cf[2]*rp[base+idx[2]] + cf[3]*rp[base+idx[3]];
        acc = fmaf(dp[base + n], r, acc);
    }
    acc += __shfl_xor(acc, 16, 32);
    acc += __shfl_xor(acc, 8, 32);
    acc += __shfl_xor(acc, 4, 32);
    acc += __shfl_xor(acc, 2, 32);
    acc += __shfl_xor(acc, 1, 32);
    if (lane == 0)
        pos_sim[(size_t)b * Nn + n] = acc * inv_dep[(size_t)b * Nn + n] * INVT;
}

// ---------------- Kernel 3a: gather normalized random-negative columns ----------------
__global__ void kgatherA(const float* __restrict__ rgb, const int* __restrict__ rand_idx,
                         const float* __restrict__ inv_rgb, float* __restrict__ A_ws) {
    int b = blockIdx.x, tid = threadIdx.x;          // 256 threads: tid == channel
    for (int k = 0; k < NRAND; ++k) {
        int idx = rand_idx[b * NRAND + k];
        float s = inv_rgb[(size_t)b * Nn + idx];
        A_ws[((size_t)b * NRAND + k) * Cc + tid] =
            rgb[((size_t)b * Cc + tid) * Nn + idx] * s;
    }
}

// ---------------- Kernel 3b: rand_neg GEMM via v_wmma_f32_16x16x4_f32 + fused LSE/max ----------------
__global__ void krandneg(const float* __restrict__ dep, const float* __restrict__ A_ws,
                         const float* __restrict__ inv_dep, const float* __restrict__ pos_sim,
                         float* __restrict__ lse_base, float* __restrict__ max_neg) {
    __shared__ float As[32 * APAD];                 // 32 negatives x 256 channels, padded
    int b = blockIdx.y;
    int tid = threadIdx.x;
    for (int i = 0; i < 32; ++i)
        As[i * APAD + tid] = A_ws[((size_t)b * NRAND + i) * Cc + tid];
    __syncthreads();

    int wave = tid >> 5, lane = tid & 31;
    int m = lane & 15;
    int koff = (lane >> 4) << 1;                    // 0 or 2 (lane-half selects K pair)
    int gwave = blockIdx.x * 8 + wave;              // 0..255
    const float* dp = dep + (size_t)b * Cc * Nn;

    for (int it = 0; it < 4; ++it) {
        int tile = gwave * 4 + it;                  // 1024 tiles of 16 pixels
        int col = tile * 16 + (lane & 15);
        v8f acc0 = {0,0,0,0,0,0,0,0};               // negatives 0..15
        v8f acc1 = {0,0,0,0,0,0,0,0};               // negatives 16..31
        for (int c0 = 0; c0 < Cc; c0 += 4) {
            v2f bvv, a0, a1;
            bvv.x = dp[(size_t)(c0 + koff) * Nn + col];
            bvv.y = dp[(size_t)(c0 + koff + 1) * Nn + col];
            a0.x = As[m * APAD + c0 + koff];
            a0.y = As[m * APAD + c0 + koff + 1];
            a1.x = As[(m + 16) * APAD + c0 + koff];
            a1.y = As[(m + 16) * APAD + c0 + koff + 1];
            acc0 = __builtin_amdgcn_wmma_f32_16x16x4_f32(false, a0, false, bvv,
                                                         (short)0, acc0, false, false);
            acc1 = __builtin_amdgcn_wmma_f32_16x16x4_f32(false, a1, false, bvv,
                                                         (short)0, acc1, false, false);
        }
        // Epilogue: lanes l and l^16 hold the same column (rows m and m+8).
        float scale = inv_dep[(size_t)b * Nn + col] * INVT;
        float vals[16];
        #pragma unroll
        for (int j = 0; j < 8; ++j) { vals[j] = acc0[j] * scale; vals[8 + j] = acc1[j] * scale; }
        float vmax = vals[0];
        #pragma unroll
        for (int j = 1; j < 16; ++j) vmax = fmaxf(vmax, vals[j]);
        float m32 = fmaxf(vmax, __shfl_xor(vmax, 16, 32));   // max over all 32 negatives
        float pos = pos_sim[(size_t)b * Nn + col];
        float M = fmaxf(m32, pos);
        float s = 0.f;
        #pragma unroll
        for (int j = 0; j < 16; ++j) s += expf(vals[j] - M);
        float tot = s + __shfl_xor(s, 16, 32) + expf(pos - M);
        if (lane < 16) {
            lse_base[(size_t)b * Nn + col] = M + logf(tot);
            max_neg[(size_t)b * Nn + col] = m32;
        }
    }
}

// ---------------- Kernel 4: hard negatives (per grid position) ----------------
__global__ void khard(const float* __restrict__ rgb, const float* __restrict__ dep,
                      const float* __restrict__ pc,
                      const float* __restrict__ inv_rgb, const float* __restrict__ inv_dep,
                      const int* __restrict__ off_u, const int* __restrict__ off_v,
                      float* __restrict__ lse_hard, float* __restrict__ mh) {
    __shared__ float dg[Cc];
    __shared__ float red[128];
    int b = blockIdx.y, p = blockIdx.x;
    int gi = p >> 4, gj = p & 15;
    int gidx = (gi * 8) * Ww + gj * 8;              // grid_idx[p]
    int tid = threadIdx.x;                          // 128 threads
    float idn = inv_dep[(size_t)b * Nn + gidx];
    dg[tid]       = dep[((size_t)b * Cc + tid)       * Nn + gidx] * idn;
    dg[tid + 128] = dep[((size_t)b * Cc + tid + 128) * Nn + gidx] * idn;
    __syncthreads();

    float sim = -INFINITY;
    if (tid < NHARD) {
        float pu = pc[((size_t)b * 2 + 0) * Nn + gidx];
        float pv = pc[((size_t)b * 2 + 1) * Nn + gidx];
        int pui = (int)fminf(fmaxf(pu, 0.f), (float)(Ww - 1));
        int pvi = (int)fminf(fmaxf(pv, 0.f), (float)(Hh - 1));
        int ou = off_u[((size_t)b * NHARD + tid) * NPOS + p];
        int ov = off_v[((size_t)b * NHARD + tid) * NPOS + p];
        if (ou == 0 && ov == 0) ou = 1;
        int hu = min(max(pui + ou, 0), Ww - 1);
        int hv = min(max(pvi + ov, 0), Hh - 1);
        int col = hv * Ww + hu;
        const float* rp = rgb + (size_t)b * Cc * Nn + col;
        float s = 0.f;
        for (int c = 0; c < Cc; ++c) s = fmaf(dg[c], rp[(size_t)c * Nn], s);
        sim = s * inv_rgb[(size_t)b * Nn + col] * INVT;
    }
    red[tid] = sim;
    __syncthreads();
    for (int st = 64; st > 0; st >>= 1) {
        if (tid < st) red[tid] = fmaxf(red[tid], red[tid + st]);
        __syncthreads();
    }
    float M = red[0];
    __syncthreads();
    red[tid] = (tid < NHARD) ? expf(sim - M) : 0.f;
    __syncthreads();
    for (int st = 64; st > 0; st >>= 1) {
        if (tid < st) red[tid] += red[tid + st];
        __syncthreads();
    }
    if (tid == 0) {
        lse_hard[b * NPOS + p] = M + logf(red[0]);
        mh[b * NPOS + p] = M;
    }
}

// ---------------- Kernel 5: per-block partial reductions (deterministic) ----------------
__global__ void kpartial(const float* __restrict__ mask, const float* __restrict__ pos_sim,
                         const float* __restrict__ lse_base, const float* __restrict__ max_neg,
                         const float* __restrict__ lse_hard, const float* __restrict__ mh,
                         float* __restrict__ partials) {
    __shared__ float s0[256], s1[256], s2[256];
    int e = blockIdx.x * 256 + threadIdx.x;         // < B*N = 131072 (512 blocks)
    int b = e / Nn, n = e % Nn;
    float mk = mask[e];
    float lb = lse_base[e];
    float mn = max_neg[e];
    float ps = pos_sim[e];
    int u = n & (Ww - 1), v = n >> 7;
    if (((u & 7) == 0) && ((v & 7) == 0)) {         // grid position
        int p = (v >> 3) * 16 + (u >> 3);
        float lh = lse_hard[b * NPOS + p];
        float M = fmaxf(lb, lh);
        lb = M + logf(expf(lb - M) + expf(lh - M)); // logaddexp
        mn = fmaxf(mn, mh[b * NPOS + p]);
    }
    s0[threadIdx.x] = mk;
    s1[threadIdx.x] = (lb - ps) * mk;
    s2[threadIdx.x] = (ps > mn && mk > 0.5f) ? 1.0f : 0.0f;
    __syncthreads();
    for (int st = 128; st > 0; st >>= 1) {
        if (threadIdx.x < st) {
            s0[threadIdx.x] += s0[threadIdx.x + st];
            s1[threadIdx.x] += s1[threadIdx.x + st];
            s2[threadIdx.x] += s2[threadIdx.x + st];
        }
        __syncthreads();
    }
    if (threadIdx.x == 0) {
        partials[blockIdx.x * 3 + 0] = s0[0];
        partials[blockIdx.x * 3 + 1] = s1[0];
        partials[blockIdx.x * 3 + 2] = s2[0];
    }
}

// ---------------- Kernel 6: final scalar outputs ----------------
__global__ void kfinal(const float* __restrict__ partials, float* __restrict__ out) {
    __shared__ float s0[256], s1[256], s2[256];
    int tid = threadIdx.x;
    float a = 0.f, l = 0.f, c = 0.f;
    for (int i = tid; i < 512; i += 256) {
        a += partials[i * 3 + 0];
        l += partials[i * 3 + 1];
        c += partials[i * 3 + 2];
    }
    s0[tid] = a; s1[tid] = l; s2[tid] = c;
    __syncthreads();
    for (int st = 128; st > 0; st >>= 1) {
        if (tid < st) {
            s0[tid] += s0[tid + st];
            s1[tid] += s1[tid + st];
            s2[tid] += s2[tid + st];
        }
        __syncthreads();
    }
    if (tid == 0) {
        float denom = fmaxf(s0[0], 1.0f);
        out[0] = s1[0] / denom;
        out[1] = s2[0] / denom * 100.0f;
    }
}

extern "C" void kernel_launch(void* const* d_in, const int* in_sizes, int n_in,
                              void* d_out, int out_size, void* d_ws, size_t ws_size,
                              hipStream_t stream) {
    (void)in_sizes; (void)n_in; (void)out_size; (void)ws_size;
    const float* rgb  = (const float*)d_in[0];
    const float* dep  = (const float*)d_in[1];
    const float* pc   = (const float*)d_in[2];
    const float* mask = (const float*)d_in[3];
    const int* rand_idx = (const int*)d_in[4];
    const int* off_u    = (const int*)d_in[5];
    const int* off_v    = (const int*)d_in[6];

    const size_t BN = (size_t)Bb * Nn;
    float* ws       = (float*)d_ws;
    float* inv_rgb  = ws;
    float* inv_dep  = inv_rgb + BN;
    float* pos_sim  = inv_dep + BN;
    float* lse_base = pos_sim + BN;
    float* max_neg  = lse_base + BN;
    float* A_ws     = max_neg + BN;                 // B*32*256
    float* lse_hard = A_ws + (size_t)Bb * NRAND * Cc;
    float* mhv      = lse_hard + Bb * NPOS;
    float* partials = mhv + Bb * NPOS;              // 512*3
    float* out      = (float*)d_out;

    knorm   <<<dim3((unsigned)(BN / 256)), dim3(256), 0, stream>>>(rgb, dep, inv_rgb, inv_dep);
    kpos    <<<dim3((unsigned)(BN / 8)),   dim3(256), 0, stream>>>(rgb, dep, pc, inv_rgb, inv_dep, pos_sim);
    kgatherA<<<dim3(Bb),                   dim3(256), 0, stream>>>(rgb, rand_idx, inv_rgb, A_ws);
    krandneg<<<dim3(32, Bb),               dim3(256), 0, stream>>>(dep, A_ws, inv_dep, pos_sim, lse_base, max_neg);
    khard   <<<dim3(NPOS, Bb),             dim3(128), 0, stream>>>(rgb, dep, pc, inv_rgb, inv_dep, off_u, off_v, lse_hard, mhv);
    kpartial<<<dim3(512),                  dim3(256), 0, stream>>>(mask, pos_sim, lse_base, max_neg, lse_hard, mhv, partials);
    kfinal  <<<dim3(1),                    dim3(256), 0, stream>>>(partials, out);
}